// ThirawatGRUNet_78761110274471
// MI455X (gfx1250) — compile-verified
//
#include <hip/hip_runtime.h>
#include <math.h>

#define Lx 8
#define Hx 2048
#define Ox 1000

typedef __attribute__((ext_vector_type(2))) float v2f;
typedef __attribute__((ext_vector_type(8))) float v8f;

// ---------------------------------------------------------------------------
// Kernel 1: 48 independent GEMVs  y[i] = W_i @ v_i   (bias folded in later)
// grid = (16 row-tiles, 48 matrices), block = 256 (8 wave32)
// Each wave computes 16 output rows with V_WMMA_F32_16X16X4_F32.
// ---------------------------------------------------------------------------
__global__ __launch_bounds__(256) void gemv48_kernel(
    const float* __restrict__ x, const float* __restrict__ h,
    const float* __restrict__ W_ir, const float* __restrict__ W_hr,
    const float* __restrict__ W_h1, const float* __restrict__ W_x1,
    const float* __restrict__ W_iu, const float* __restrict__ W_hu,
    float* __restrict__ G)
{
    __shared__ float vec[Hx];

    const int l = blockIdx.y & 7;   // layer
    const int g = blockIdx.y >> 3;  // gate: 0=ir 1=hr 2=h1 3=x1 4=iu 5=hu

    const float* Wbase;
    switch (g) {
        case 0: Wbase = W_ir; break;
        case 1: Wbase = W_hr; break;
        case 2: Wbase = W_h1; break;
        case 3: Wbase = W_x1; break;
        case 4: Wbase = W_iu; break;
        default: Wbase = W_hu; break;
    }
    const float* W = Wbase + (size_t)l * Hx * Hx;

    // gates ir / x1 / iu consume inp[l] = (l==0 ? x : h[l-1]); hr / h1 / hu consume h[l]
    const bool useInp = (g == 0) | (g == 3) | (g == 4);
    const float* v = useInp ? (l == 0 ? x : h + (size_t)(l - 1) * Hx)
                            : h + (size_t)l * Hx;

    // stage the input vector in LDS (broadcast reads in the hot loop)
    for (int i = threadIdx.x; i < Hx; i += 256) vec[i] = v[i];
    __syncthreads();

    const int wave = threadIdx.x >> 5;
    const int lane = threadIdx.x & 31;
    const int half = lane >> 4;                       // lane half selects K-slot pair
    const int rowBase = blockIdx.x * 128 + wave * 16; // 16 rows per wave
    const float* wrow = W + (size_t)(rowBase + (lane & 15)) * Hx;

    v8f c = {};
    #pragma unroll 4
    for (int k = 0; k < Hx; k += 8) {
        // contiguous 16B per lane; halves cover k..k+3 and k+4..k+7 of the row
        const float4 w4 = *(const float4*)(wrow + k + 4 * half);
        // A-frag slot layout (16x4 f32): VGPR0 = slots {0 | 2}, VGPR1 = slots {1 | 3}
        // WMMA #0 uses K set {k, k+1 | k+4, k+5}; WMMA #1 uses {k+2, k+3 | k+6, k+7}
        v2f a0; a0.x = w4.x; a0.y = w4.y;
        v2f a1; a1.x = w4.z; a1.y = w4.w;
        // B broadcast: B[kappa][n] = vec[kappa] for all 16 columns n
        const v2f b0 = *(const v2f*)(vec + k + 4 * half);
        const v2f b1 = *(const v2f*)(vec + k + 4 * half + 2);
        c = __builtin_amdgcn_wmma_f32_16x16x4_f32(false, a0, false, b0,
                                                  (short)0, c, false, false);
        c = __builtin_amdgcn_wmma_f32_16x16x4_f32(false, a1, false, b1,
                                                  (short)0, c, false, false);
    }

    // D layout: VGPR r holds M=r (lanes 0-15) and M=8+r (lanes 16-31); all
    // columns are identical, so lane 0 / lane 16 each hold 8 row results.
    if ((lane & 15) == 0) {
        float* dst = G + (size_t)blockIdx.y * Hx + rowBase + 8 * half;
        #pragma unroll
        for (int r = 0; r < 8; ++r) dst[r] = c[r];
    }
}

// ---------------------------------------------------------------------------
// Kernel 2: gate math + bias fold -> new_h  (written straight into d_out)
// ---------------------------------------------------------------------------
__device__ __forceinline__ float sigmoidf_(float z) {
    return 1.0f / (1.0f + expf(-z));
}

__global__ __launch_bounds__(256) void combine_kernel(
    const float* __restrict__ G, const float* __restrict__ h,
    const float* __restrict__ b_ir, const float* __restrict__ b_hr,
    const float* __restrict__ b_h1, const float* __restrict__ b_x1,
    const float* __restrict__ b_iu, const float* __restrict__ b_hu,
    float* __restrict__ newh)
{
    const int idx = blockIdx.x * 256 + threadIdx.x;  // < 8*2048
    const int l = idx >> 11;
    const int j = idx & (Hx - 1);
    const int bj = l * Hx + j;

    const float gir = G[(0 * Lx + l) * Hx + j] + b_ir[bj];
    const float ghr = G[(1 * Lx + l) * Hx + j] + b_hr[bj];
    const float gh1 = G[(2 * Lx + l) * Hx + j] + b_h1[bj];
    const float gx1 = G[(3 * Lx + l) * Hx + j] + b_x1[bj];
    const float giu = G[(4 * Lx + l) * Hx + j] + b_iu[bj];
    const float ghu = G[(5 * Lx + l) * Hx + j] + b_hu[bj];

    const float reset  = sigmoidf_(gir + ghr);
    const float r      = tanhf(reset * gh1 + gx1);
    const float update = sigmoidf_(giu + ghu);
    newh[idx] = (1.0f - update) * r + update * h[idx];
}

// ---------------------------------------------------------------------------
// Kernel 3: logits[o] = relu(new_h[7]) . W_fc[o] + b_fc[o]   (1 wave / row)
// ---------------------------------------------------------------------------
__global__ __launch_bounds__(256) void fc_kernel(
    const float* __restrict__ h7, const float* __restrict__ Wfc,
    const float* __restrict__ bfc, float* __restrict__ logits)
{
    const int o = blockIdx.x * 8 + (threadIdx.x >> 5);  // 125*8 = 1000 exactly
    const int lane = threadIdx.x & 31;
    const float* wr = Wfc + (size_t)o * Hx;

    float s = 0.0f;
    for (int k = lane; k < Hx; k += 32)
        s += fmaxf(h7[k], 0.0f) * wr[k];
    #pragma unroll
    for (int off = 16; off > 0; off >>= 1)
        s += __shfl_xor(s, off, 32);
    if (lane == 0) logits[o] = s + bfc[o];
}

// ---------------------------------------------------------------------------
// Kernel 4: log_softmax over 1000 logits (single block)
// ---------------------------------------------------------------------------
__global__ __launch_bounds__(256) void lsm_kernel(
    const float* __restrict__ logits, float* __restrict__ out)
{
    __shared__ float red[256];
    const int t = threadIdx.x;

    float m = -3.402823466e38f;
    for (int i = t; i < Ox; i += 256) m = fmaxf(m, logits[i]);
    red[t] = m; __syncthreads();
    for (int s = 128; s > 0; s >>= 1) {
        if (t < s) red[t] = fmaxf(red[t], red[t + s]);
        __syncthreads();
    }
    const float gmax = red[0];
    __syncthreads();

    float sum = 0.0f;
    for (int i = t; i < Ox; i += 256) sum += expf(logits[i] - gmax);
    red[t] = sum; __syncthreads();
    for (int s = 128; s > 0; s >>= 1) {
        if (t < s) red[t] += red[t + s];
        __syncthreads();
    }
    const float lse = gmax + logf(red[0]);
    for (int i = t; i < Ox; i += 256) out[i] = logits[i] - lse;
}

// ---------------------------------------------------------------------------
extern "C" void kernel_launch(void* const* d_in, const int* in_sizes, int n_in,
                              void* d_out, int out_size, void* d_ws, size_t ws_size,
                              hipStream_t stream) {
    const float* x    = (const float*)d_in[0];
    const float* h    = (const float*)d_in[1];
    const float* W_ir = (const float*)d_in[2];
    const float* b_ir = (const float*)d_in[3];
    const float* W_hr = (const float*)d_in[4];
    const float* b_hr = (const float*)d_in[5];
    const float* W_h1 = (const float*)d_in[6];
    const float* b_h1 = (const float*)d_in[7];
    const float* W_x1 = (const float*)d_in[8];
    const float* b_x1 = (const float*)d_in[9];
    const float* W_iu = (const float*)d_in[10];
    const float* b_iu = (const float*)d_in[11];
    const float* W_hu = (const float*)d_in[12];
    const float* b_hu = (const float*)d_in[13];
    const float* W_fc = (const float*)d_in[14];
    const float* b_fc = (const float*)d_in[15];

    float* out    = (float*)d_out;   // [1000] log_softmax result
    float* newh   = out + Ox;        // [8*2048] new_h (second tuple element)
    float* G      = (float*)d_ws;    // [48*2048] raw matvec results
    float* logits = G + 48 * Hx;     // [1000]

    dim3 g1(16, 48);
    gemv48_kernel<<<g1, 256, 0, stream>>>(x, h, W_ir, W_hr, W_h1, W_x1, W_iu, W_hu, G);
    combine_kernel<<<64, 256, 0, stream>>>(G, h, b_ir, b_hr, b_h1, b_x1, b_iu, b_hu, newh);
    fc_kernel<<<125, 256, 0, stream>>>(newh + 7 * Hx, W_fc, b_fc, logits);
    lsm_kernel<<<1, 256, 0, stream>>>(logits, out);
}